// pAUCLoss_17197049053489
// MI455X (gfx1250) — compile-verified
//
#include <hip/hip_runtime.h>

#define N_NEG 16384
#define N_POS 8192
#define TOPK  512
#define EPSF  1e-6f

typedef float v2f __attribute__((ext_vector_type(2)));
typedef float v8f __attribute__((ext_vector_type(8)));

// ---------------------------------------------------------------------------
// Kernel 1: full bitonic sort (descending) of the 16384 negatives in LDS,
// then emit the top 512 values. One workgroup, 1024 threads, 64 KB LDS.
// ---------------------------------------------------------------------------
__global__ __launch_bounds__(1024)
void topk_sort_kernel(const float* __restrict__ neg, float* __restrict__ topneg) {
    __shared__ float s[N_NEG];           // 65536 bytes (WGP has 320 KB LDS)
    const unsigned tid = threadIdx.x;

    for (unsigned i = tid; i < N_NEG; i += 1024u) s[i] = neg[i];
    __syncthreads();

    for (unsigned k = 2; k <= N_NEG; k <<= 1) {
        for (unsigned j = k >> 1; j > 0; j >>= 1) {
            for (unsigned i = tid; i < N_NEG; i += 1024u) {
                unsigned ixj = i ^ j;
                if (ixj > i) {           // each pair handled exactly once
                    float a = s[i], b = s[ixj];
                    bool desc = ((i & k) == 0);          // this block: descending
                    if (desc ? (a < b) : (a > b)) { s[i] = b; s[ixj] = a; }
                }
            }
            __syncthreads();
        }
    }
    for (unsigned i = tid; i < TOPK; i += 1024u) topneg[i] = s[i];
}

// ---------------------------------------------------------------------------
// Kernel 2: each wave handles 16 positive rows x all 512 top negatives.
// One V_WMMA_F32_16X16X4_F32 per 16x16 tile builds yy = pos_m - neg_n:
//   A[m,:] = [pos_m, 1, 0, 0]   (16x4)   B[:,n] = [1, -neg_n, 0, 0]^T (4x16)
// A layout (32-bit A 16x4): lanes 0-15 -> K=0,1 in v[0],v[1]; lanes 16-31 -> K=2,3.
// B layout (4x16): lanes 0-15 hold N, v[0]=K0 / v[1]=K1; lanes 16-31 -> K=2,3.
// Then bce = -log(clip(sigmoid(yy), eps, 1-eps)) elementwise on the 8
// accumulator VGPRs, accumulated per lane, wave-reduced via shuffles.
// ---------------------------------------------------------------------------
__global__ __launch_bounds__(256)
void pauc_wmma_kernel(const float* __restrict__ pos,
                      const float* __restrict__ topneg,
                      float* __restrict__ partials) {
    const int lane = threadIdx.x & 31;
    const int wave = (int)(blockIdx.x * 256u + threadIdx.x) >> 5;   // 0..511
    const int m0   = wave * 16;

    v2f a;
    a.x = (lane < 16) ? pos[m0 + lane] : 0.0f;   // K=0 : pos_m   (lanes 16-31: K=2 -> 0)
    a.y = (lane < 16) ? 1.0f           : 0.0f;   // K=1 : 1       (lanes 16-31: K=3 -> 0)

    float acc = 0.0f;

    for (int nt = 0; nt < TOPK / 16; ++nt) {
        v2f b;
        b.x = (lane < 16) ? 1.0f : 0.0f;                       // K=0 row: ones
        b.y = (lane < 16) ? -topneg[nt * 16 + lane] : 0.0f;    // K=1 row: -neg_n
        v8f c = {};
        // D[m][n] = pos_m*1 + 1*(-neg_n) = pos_m - neg_n (exact products, one RNE add)
        v8f d = __builtin_amdgcn_wmma_f32_16x16x4_f32(
            /*neg_a=*/false, a, /*neg_b=*/false, b,
            /*c_mod=*/(short)0, c, /*reuse_a=*/false, /*reuse_b=*/false);

#pragma unroll
        for (int r = 0; r < 8; ++r) {
            float yy = d[r];
            float p  = 1.0f / (1.0f + expf(-yy));
            p = fminf(fmaxf(p, EPSF), 1.0f - EPSF);
            acc -= logf(p);
        }
    }

    // wave32 reduction
#pragma unroll
    for (int off = 16; off > 0; off >>= 1)
        acc += __shfl_down(acc, off, 32);
    if (lane == 0) partials[wave] = acc;
}

// ---------------------------------------------------------------------------
// Kernel 3: deterministic fixed-order final reduction + scaling.
// ---------------------------------------------------------------------------
__global__ void finalize_kernel(const float* __restrict__ partials,
                                float* __restrict__ out) {
    if (threadIdx.x == 0 && blockIdx.x == 0) {
        double s = 0.0;
        for (int i = 0; i < N_POS / 16; ++i) s += (double)partials[i];
        const double denom = (double)N_POS * (double)N_NEG;   // 2^27
        out[0] = (float)(s / denom);
    }
}

// ---------------------------------------------------------------------------
extern "C" void kernel_launch(void* const* d_in, const int* in_sizes, int n_in,
                              void* d_out, int out_size, void* d_ws, size_t ws_size,
                              hipStream_t stream) {
    const float* score_neg = (const float*)d_in[0];   // 16384
    const float* score_pos = (const float*)d_in[1];   // 8192

    float* topneg   = (float*)d_ws;          // [0 .. 511]
    float* partials = topneg + TOPK;         // [512 .. 1023]

    topk_sort_kernel<<<1, 1024, 0, stream>>>(score_neg, topneg);
    pauc_wmma_kernel<<<(N_POS / 16) * 32 / 256, 256, 0, stream>>>(score_pos, topneg, partials);
    finalize_kernel<<<1, 32, 0, stream>>>(partials, (float*)d_out);
}